// BSScanThru_39298950758456
// MI455X (gfx1250) — compile-verified
//
#include <hip/hip_runtime.h>
#include <cstdint>

// ---------------------------------------------------------------------------
// BSScanThru for MI455X (gfx1250, wave32).
// Single-pass decoupled-lookback (g,p) carry scan, 12 B/elem -> ~35us roofline.
// Data movement uses CDNA5 async global->LDS copies; output uses NT stores.
// ---------------------------------------------------------------------------

#define TILE_ELEMS 4096
#define BLOCK_THREADS 256
#define ITEMS 16

typedef __attribute__((ext_vector_type(4))) float v4f;

static __device__ __forceinline__ unsigned brev8(unsigned v) {
  return __builtin_bitreverse32(v) >> 24;
}

static __device__ __forceinline__ unsigned ballot32(int p) {
#if __has_builtin(__builtin_amdgcn_ballot_w32)
  return __builtin_amdgcn_ballot_w32(p != 0);
#else
  return (unsigned)__ballot(p);
#endif
}

// ---- CDNA5 async global->LDS copy (guarded) -------------------------------
#if __has_builtin(__builtin_amdgcn_global_load_async_to_lds_b128)
#define BS_ASYNC_LDS 1
typedef __attribute__((ext_vector_type(4))) int v4i;
typedef __attribute__((address_space(1))) v4i g_v4i;
typedef __attribute__((address_space(3))) v4i l_v4i;
static __device__ __forceinline__ void async_b128(const void* gsrc, void* ldst) {
  g_v4i* g = (g_v4i*)(unsigned long long)(uintptr_t)gsrc;
  l_v4i* l = (l_v4i*)(unsigned)(uintptr_t)ldst;  // low 32 bits = LDS offset
  __builtin_amdgcn_global_load_async_to_lds_b128(g, l, 0, 0);
}
#else
#define BS_ASYNC_LDS 0
#endif

static __device__ __forceinline__ void wait_async0() {
#if __has_builtin(__builtin_amdgcn_s_wait_asynccnt)
  __builtin_amdgcn_s_wait_asynccnt(0);
#else
  asm volatile("s_wait_asynccnt 0" ::: "memory");
#endif
  asm volatile("" ::: "memory");  // compiler barrier: LDS reads stay below
}

// ---------------------------------------------------------------------------
__global__ void bs_clear_ws_kernel(unsigned* __restrict__ ws, int n) {
  int i = blockIdx.x * blockDim.x + threadIdx.x;
  if (i < n) ws[i] = 0u;
}

__global__ __launch_bounds__(BLOCK_THREADS) void bs_scanthru_kernel(
    const int* __restrict__ a_, const int* __restrict__ b_,
    float* __restrict__ out, unsigned* __restrict__ ws, int n) {
  const unsigned* a = (const unsigned*)a_;
  const unsigned* b = (const unsigned*)b_;
  unsigned* desc = ws + 16;

  __shared__ __align__(16) unsigned aTile[TILE_ELEMS];
  __shared__ __align__(16) unsigned bTile[TILE_ELEMS];
  __shared__ unsigned waveG[8], waveP[8], waveCin[8];
  __shared__ unsigned sTile;

  // --- ordered tile assignment (deterministic; lookback-safe) ---
  if (threadIdx.x == 0)
    sTile = __hip_atomic_fetch_add(&ws[0], 1u, __ATOMIC_RELAXED,
                                   __HIP_MEMORY_SCOPE_AGENT);
  __syncthreads();
  const unsigned tile = sTile;
  const int tileBase = (int)tile * TILE_ELEMS;
  const int t16 = (int)threadIdx.x * ITEMS;
  const int lane = threadIdx.x & 31;
  const int wv = threadIdx.x >> 5;
  const bool fullTile = (tileBase + TILE_ELEMS) <= n;

  // --- stage tile into LDS ---
  if (fullTile) {
#if BS_ASYNC_LDS
    // lane-contiguous 512B bursts; each wave loads exactly the region its own
    // lanes consume, so s_wait_asynccnt 0 (per-wave) is the only sync needed.
    const unsigned* ga = a + tileBase;
    const unsigned* gb = b + tileBase;
#pragma unroll
    for (int k = 0; k < 4; ++k) {
      int c = wv * 128 + k * 32 + lane;  // 16B chunk index within tile
      async_b128(ga + c * 4, &aTile[c * 4]);
      async_b128(gb + c * 4, &bTile[c * 4]);
    }
    wait_async0();
#else
    const uint4* ga4 = (const uint4*)(a + tileBase);
    const uint4* gb4 = (const uint4*)(b + tileBase);
#pragma unroll
    for (int q = 0; q < 4; ++q) {
      *(uint4*)&aTile[t16 + q * 4] = ga4[(int)threadIdx.x * 4 + q];
      *(uint4*)&bTile[t16 + q * 4] = gb4[(int)threadIdx.x * 4 + q];
    }
#endif
  } else {
#pragma unroll
    for (int j = 0; j < ITEMS; ++j) {
      int idx = tileBase + t16 + j;
      aTile[t16 + j] = (idx < n) ? a[idx] : 0u;
      bTile[t16 + j] = (idx < n) ? b[idx] : 0u;
    }
  }

  // --- phase A: per-item (c,g,p); pack into registers ---
  unsigned gm = 0, pm = 0;
  unsigned cpack[4], npack[4];
#pragma unroll
  for (int q = 0; q < 4; ++q) {
    uint4 av = *(const uint4*)&aTile[t16 + q * 4];
    uint4 bv = *(const uint4*)&bTile[t16 + q * 4];
    unsigned cw = 0, nw = 0;
#define BS_PROC(jq, AV, BV)                              \
  {                                                      \
    unsigned ar = brev8(AV), br_ = brev8(BV);            \
    unsigned c = (ar + br_) & 0xFFu;                     \
    unsigned jj = (unsigned)(q * 4 + (jq));              \
    gm |= (unsigned)(c < ar) << jj;                      \
    pm |= (unsigned)(c == 0xFFu) << jj;                  \
    cw |= c << (8 * (jq));                               \
    nw |= ((~(BV)) & 0xFFu) << (8 * (jq));               \
  }
    BS_PROC(0, av.x, bv.x)
    BS_PROC(1, av.y, bv.y)
    BS_PROC(2, av.z, bv.z)
    BS_PROC(3, av.w, bv.w)
#undef BS_PROC
    cpack[q] = cw;
    npack[q] = nw;
  }

  // thread aggregate via 16-bit adder-carry trick (g & p disjoint)
  unsigned Gt = (((gm | pm) + gm) >> 16) & 1u;
  unsigned Pt = (pm == 0xFFFFu) ? 1u : 0u;

  // wave aggregate via ballot + 32-bit adder-carry trick
  unsigned x = ballot32((int)Gt);
  unsigned y = ballot32((int)Pt);
  if (lane == 0) {
    unsigned long long s0 = (unsigned long long)(x | y) + x;
    waveG[wv] = (unsigned)(s0 >> 32) & 1u;
    waveP[wv] = (y == 0xFFFFFFFFu) ? 1u : 0u;
  }
  __syncthreads();

  // --- thread 0: block aggregate, decoupled lookback, per-wave carry-ins ---
  if (threadIdx.x == 0) {
    unsigned Gb = 0, Pb = 1;
#pragma unroll
    for (int w = 0; w < 8; ++w) {
      Gb = waveG[w] | (waveP[w] & Gb);
      Pb &= waveP[w];
    }
    // publish AGGREGATE early so successors can progress
    __hip_atomic_store(&desc[tile], 1u | (Gb << 2) | (Pb << 3),
                       __ATOMIC_RELEASE, __HIP_MEMORY_SCOPE_AGENT);
    unsigned cin = 0;
    if (tile != 0) {
      unsigned Gacc = 0, Pacc = 1;
      int k = (int)tile - 1;
      for (;;) {
        unsigned d = __hip_atomic_load(&desc[k], __ATOMIC_ACQUIRE,
                                       __HIP_MEMORY_SCOPE_AGENT);
        unsigned st = d & 3u;
        if (st == 0u) { __builtin_amdgcn_s_sleep(1); continue; }
        unsigned dg = (d >> 2) & 1u, dp = (d >> 3) & 1u;
        if (st == 2u) { cin = Gacc | (Pacc & dg); break; }  // PREFIX
        Gacc = Gacc | (Pacc & dg);                          // AGGREGATE
        Pacc &= dp;
        if (!Pacc) { cin = Gacc; break; }  // carry no longer depends further left
        if (--k < 0) { cin = Gacc; break; }
      }
    }
    unsigned Gout = Gb | (Pb & cin);  // inclusive carry-out of tiles [0..tile]
    __hip_atomic_store(&desc[tile], 2u | (Gout << 2), __ATOMIC_RELEASE,
                       __HIP_MEMORY_SCOPE_AGENT);
    unsigned cur = cin;
#pragma unroll
    for (int w = 0; w < 8; ++w) {
      waveCin[w] = cur;
      cur = waveG[w] | (waveP[w] & cur);
    }
  }
  __syncthreads();

  // --- phase B: resolve carries, finish brev + AND, store ---
  unsigned wcin = waveCin[wv];
  unsigned long long sa = (unsigned long long)(x | y) + x + wcin;
  unsigned tcin = (((unsigned)sa ^ y) >> lane) & 1u;
  unsigned s16 = (gm | pm) + gm + tcin;
  unsigned ic = (s16 ^ pm) & 0xFFFFu;  // per-item carry-in bits

#pragma unroll
  for (int q = 0; q < 4; ++q) {
    v4f f;
#define BS_OUT(jq, DST)                                              \
  {                                                                  \
    unsigned jj = (unsigned)(q * 4 + (jq));                          \
    unsigned c = (cpack[q] >> (8 * (jq))) & 0xFFu;                   \
    unsigned res = (c + ((ic >> jj) & 1u)) & 0xFFu;                  \
    DST = (float)(brev8(res) & ((npack[q] >> (8 * (jq))) & 0xFFu));  \
  }
    BS_OUT(0, f.x)
    BS_OUT(1, f.y)
    BS_OUT(2, f.z)
    BS_OUT(3, f.w)
#undef BS_OUT
    if (fullTile) {
      // streamed once, never re-read: non-temporal keeps 192MB L2 for inputs
      __builtin_nontemporal_store(f, (v4f*)(out + tileBase + t16) + q);
    } else {
      int base = tileBase + t16 + q * 4;
      if (base + 0 < n) out[base + 0] = f.x;
      if (base + 1 < n) out[base + 1] = f.y;
      if (base + 2 < n) out[base + 2] = f.z;
      if (base + 3 < n) out[base + 3] = f.w;
    }
  }
}

extern "C" void kernel_launch(void* const* d_in, const int* in_sizes, int n_in,
                              void* d_out, int out_size, void* d_ws,
                              size_t ws_size, hipStream_t stream) {
  (void)n_in; (void)out_size; (void)ws_size;
  const int n = in_sizes[0];
  const int numTiles = (n + TILE_ELEMS - 1) / TILE_ELEMS;
  unsigned* ws = (unsigned*)d_ws;

  // zero ticket + lookback descriptors (required every launch / graph replay)
  const int clearN = numTiles + 16;
  bs_clear_ws_kernel<<<(clearN + 255) / 256, 256, 0, stream>>>(ws, clearN);

  bs_scanthru_kernel<<<numTiles, BLOCK_THREADS, 0, stream>>>(
      (const int*)d_in[0], (const int*)d_in[1], (float*)d_out, ws, n);
}